// UniPhyBlock_28467043238292
// MI455X (gfx1250) — compile-verified
//
#include <hip/hip_runtime.h>
#include <hip/hip_bf16.h>
#include <math.h>

typedef __attribute__((ext_vector_type(2))) float v2f;
typedef __attribute__((ext_vector_type(8))) float v8f;

#define HW 4096

#define HAVE_ASYNC_LDS 0
#if defined(__has_builtin)
#if __has_builtin(__builtin_amdgcn_global_load_async_to_lds_b32)
#undef HAVE_ASYNC_LDS
#define HAVE_ASYNC_LDS 1
#endif
#endif

typedef __attribute__((address_space(1))) int* gptr_i32;
typedef __attribute__((address_space(3))) int* lptr_i32;

__device__ __forceinline__ v8f wmma4(v2f a, v2f b, v8f c) {
  // D = A(16x4) * B(4x16) + C, fp32 WMMA (VOP3P op 93)
  return __builtin_amdgcn_wmma_f32_16x16x4_f32(false, a, false, b, (short)0, c, false, false);
}

__device__ __forceinline__ v8f v8zero() {
  v8f z = {0.f,0.f,0.f,0.f,0.f,0.f,0.f,0.f};
  return z;
}

// Stage one f32 from global to LDS (async-to-LDS on CDNA5 when available).
__device__ __forceinline__ void stage_f32(const float* __restrict__ g, float* l) {
#if HAVE_ASYNC_LDS
  __builtin_amdgcn_global_load_async_to_lds_b32(
      (gptr_i32)(unsigned long long)(uintptr_t)g,
      (lptr_i32)(unsigned)(uintptr_t)l,
      0, 0);
#else
  *l = *g;
#endif
}

__device__ __forceinline__ void stage_fence() {
#if HAVE_ASYNC_LDS
  asm volatile("s_wait_asynccnt 0" ::: "memory");
#endif
  __syncthreads();
}

// ---------------------------------------------------------------- layernorm
__global__ void ln_kernel(const float* __restrict__ xr, const float* __restrict__ xi,
                          const float* __restrict__ g, const float* __restrict__ b,
                          float* __restrict__ xn) {
  int pos = blockIdx.x * blockDim.x + threadIdx.x;   // 131072 positions
  int n = pos >> 12; int p = pos & 4095;
  const float* br = xr + (size_t)n * 64 * HW + p;
  const float* bi = xi + (size_t)n * 64 * HW + p;
  float s = 0.f, s2 = 0.f;
  for (int d = 0; d < 64; ++d) { float v = br[(size_t)d * HW]; s += v; s2 += v * v; }
  for (int d = 0; d < 64; ++d) { float v = bi[(size_t)d * HW]; s += v; s2 += v * v; }
  float mu = s * (1.f / 128.f);
  float var = s2 * (1.f / 128.f) - mu * mu;
  float inv = rsqrtf(var + 1e-5f);
  float* o = xn + (size_t)n * 128 * HW + p;
  for (int c = 0; c < 128; ++c) {
    float v = (c < 64) ? br[(size_t)c * HW] : bi[(size_t)(c - 64) * HW];
    o[(size_t)c * HW] = (v - mu) * inv * g[c] + b[c];
  }
}

// --------------------------------- repack conv weights OIHW -> [tap][oc][ic]
__global__ void repack_kernel(const float* __restrict__ cw, float* __restrict__ cwt) {
  int idx = blockIdx.x * 256 + threadIdx.x;   // 147456 = 128*128*9
  int tap = idx % 9; int oi = idx / 9;        // oi = o*128 + i
  cwt[tap * 16384 + oi] = cw[idx];
}

// ------------------------------------------------- conv3x3 as 9-tap implicit GEMM
// grid: 32 images * 64 rows; block: 256 (8 waves). Wave tile: 16 x-positions x 16 out-ch.
__global__ void conv_kernel(const float* __restrict__ xn, const float* __restrict__ cwt,
                            float* __restrict__ xs) {
  int tid = threadIdx.x; int lane = tid & 31; int w = tid >> 5;
  int blk = blockIdx.x;
  int n = blk >> 6; int y = blk & 63;
  int row = lane & 15; int kh = lane >> 4;
  int mt = w & 3;
  int xm = mt * 16 + row;                 // output x (A-matrix row M)
  const float* xnb = xn + (size_t)n * 128 * HW;
  for (int i = 0; i < 4; ++i) {
    int nt = (w >> 2) * 4 + i;            // out-channel tile 0..7
    int ocol = nt * 16 + row;             // B-matrix column N
    v8f acc = v8zero();
    for (int tap = 0; tap < 9; ++tap) {
      int ky = tap / 3, kx = tap % 3;
      int yy = y + ky - 1;
      if (yy < 0 || yy > 63) continue;    // wave-uniform skip
      int xx = xm + kx - 1;
      bool va = (xx >= 0) && (xx < 64);
      int xc = xx < 0 ? 0 : (xx > 63 ? 63 : xx);   // clamp -> unconditional load
      const float* ap = xnb + yy * 64 + xc;
      const float* bp = cwt + (size_t)tap * 16384 + (size_t)ocol * 128;
      for (int k = 0; k < 128; k += 4) {
        int ic0 = k + 2 * kh;
        float a0 = ap[(size_t)ic0 * HW];
        float a1 = ap[(size_t)(ic0 + 1) * HW];
        v2f a, bb;
        a.x = va ? a0 : 0.f;              // branchless v_cndmask
        a.y = va ? a1 : 0.f;
        bb.x = bp[ic0];                   // contiguous pair -> b64
        bb.y = bp[ic0 + 1];
        acc = wmma4(a, bb, acc);
      }
    }
    float* op = xs + ((size_t)n * 128 + (size_t)nt * 16 + row) * HW + y * 64 + mt * 16;
    #pragma unroll
    for (int r = 0; r < 8; ++r) op[r + 8 * kh] = acc[r];
  }
}

// -------------------------------- xsp = x + (xs+b)*metric  (channel-last output)
__global__ void xsp_kernel(const float* __restrict__ xr, const float* __restrict__ xi,
                           const float* __restrict__ xs, const float* __restrict__ cb,
                           const float* __restrict__ metric,
                           float* __restrict__ xspr, float* __restrict__ xspi) {
  size_t idx = (size_t)blockIdx.x * blockDim.x + threadIdx.x;   // 8,388,608 in (pm,d)
  int d = (int)(idx & 63); size_t pm = idx >> 6;
  int p = (int)(pm & 4095); size_t n = pm >> 12;
  float m = metric[p];
  size_t xidx = ((size_t)n * 64 + d) * HW + p;
  size_t ir = ((size_t)n * 128 + d) * HW + p;
  size_t ii = ((size_t)n * 128 + 64 + d) * HW + p;
  xspr[idx] = xr[xidx] + (xs[ir] + cb[d]) * m;
  xspi[idx] = xi[xidx] + (xs[ii] + cb[d + 64]) * m;
}

// ------------------------------------------- complex GEMM: x_eig = xsp([pm,d]) @ E
__global__ void eig_kernel(const float* __restrict__ xsr, const float* __restrict__ xsi,
                           const float* __restrict__ Er, const float* __restrict__ Ei,
                           float* __restrict__ outr, float* __restrict__ outi) {
  __shared__ float sEr[64][66];
  __shared__ float sEi[64][66];
  int tid = threadIdx.x; int lane = tid & 31; int w = tid >> 5;
  int row = lane & 15, kh = lane >> 4;
  // stage E^T into LDS: sE[e][d] = E[d*64+e]
  for (int i = 0; i < 16; ++i) {
    int idx = i * 256 + tid;          // d*64 + e
    int d = idx >> 6, e = idx & 63;
    stage_f32(Er + idx, &sEr[e][d]);
    stage_f32(Ei + idx, &sEi[e][d]);
  }
  stage_fence();
  int nt = w >> 1;
  int col = nt * 16 + row;
  for (int s = 0; s < 4; ++s) {
    int mt = blockIdx.x * 8 + (w & 1) + s * 2;
    int gm0 = mt * 16;
    const float* arb = xsr + (size_t)(gm0 + row) * 64;
    const float* aib = xsi + (size_t)(gm0 + row) * 64;
    v8f P1 = v8zero(), P2 = v8zero(), P3 = v8zero(), P4 = v8zero();
    for (int k = 0; k < 64; k += 4) {
      int d0 = k + 2 * kh;
      v2f ar, ai, er, ei;
      ar.x = arb[d0];       ar.y = arb[d0 + 1];       // contiguous -> b64
      ai.x = aib[d0];       ai.y = aib[d0 + 1];
      er.x = sEr[col][d0];  er.y = sEr[col][d0 + 1];  // ds_load_b64
      ei.x = sEi[col][d0];  ei.y = sEi[col][d0 + 1];
      P1 = wmma4(ar, er, P1);
      P2 = wmma4(ai, ei, P2);
      P3 = wmma4(ar, ei, P3);
      P4 = wmma4(ai, er, P4);
    }
    #pragma unroll
    for (int r = 0; r < 8; ++r) {
      int m = r + 8 * kh;
      size_t o = (size_t)(gm0 + m) * 64 + col;
      outr[o] = P1[r] - P2[r];
      outi[o] = P3[r] + P4[r];
    }
  }
}

// ------------------------------------------------------- mean over H,W per (bt,e)
__global__ void mean_kernel(const float* __restrict__ er, const float* __restrict__ ei,
                            float* __restrict__ mr, float* __restrict__ mi) {
  __shared__ float sr[4][64], si[4][64];
  int bt = blockIdx.x; int tid = threadIdx.x; int e = tid & 63, sl = tid >> 6;
  const float* pr = er + (size_t)bt * HW * 64 + e;
  const float* pi = ei + (size_t)bt * HW * 64 + e;
  float ar = 0.f, ai = 0.f;
  for (int p = sl * 1024; p < sl * 1024 + 1024; ++p) {
    ar += pr[(size_t)p * 64]; ai += pi[(size_t)p * 64];
  }
  sr[sl][e] = ar; si[sl][e] = ai;
  __syncthreads();
  if (sl == 0) {
    mr[bt * 64 + e] = (sr[0][e] + sr[1][e] + sr[2][e] + sr[3][e]) * (1.f / 4096.f);
    mi[bt * 64 + e] = (si[0][e] + si[1][e] + si[2][e] + si[3][e]) * (1.f / 4096.f);
  }
}

// -------------------------------------------------------------- flux scan (T=8, tiny)
#define OUT_H_OFF   16777216
#define OUT_FLX_OFF 18874368
__global__ void flux_kernel(const float* __restrict__ mr, const float* __restrict__ mi,
                            const float* __restrict__ fpr, const float* __restrict__ fpi,
                            const float* __restrict__ dt,
                            const float* __restrict__ lfr, const float* __restrict__ lfi,
                            float* __restrict__ ffr, float* __restrict__ ffi,
                            float* __restrict__ dout) {
  int tid = threadIdx.x; int b = tid >> 6, e = tid & 63;
  float dts = dt[0];
  float lr = -fabsf(lfr[e]), li = lfi[e];
  float ea = __expf(lr * dts);
  float ar = ea * __cosf(li * dts), ai = ea * __sinf(li * dts);
  float nr = ar - 1.f, ni = ai;
  float den = 1.f / (lr * lr + li * li);
  float cr = (nr * lr + ni * li) * den, ci = (ni * lr - nr * li) * den;
  float sr = fpr[b * 64 + e], si = fpi[b * 64 + e];
  for (int t = 0; t < 8; ++t) {
    int o = (b * 8 + t) * 64 + e;
    ffr[o] = sr; ffi[o] = si;                          // flux_forcing = state before update
    float ur = mr[o] * cr - mi[o] * ci;
    float ui = mr[o] * ci + mi[o] * cr;
    float nsr = ar * sr - ai * si + ur;
    float nsi = ar * si + ai * sr + ui;
    sr = nsr; si = nsi;
  }
  dout[OUT_FLX_OFF + b * 64 + e] = sr;
  dout[OUT_FLX_OFF + 256 + b * 64 + e] = si;
}

// ------------------------------------------------------- source / gate (tiny GEMV)
__global__ void srcgate_kernel(const float* __restrict__ ffr, const float* __restrict__ ffi,
                               const float* __restrict__ wsr, const float* __restrict__ wsi,
                               const float* __restrict__ wg, const float* __restrict__ bg,
                               float* __restrict__ gq, float* __restrict__ srcr,
                               float* __restrict__ srci) {
  int gid = blockIdx.x * blockDim.x + threadIdx.x;     // 2048
  int e = gid & 63; int bt = gid >> 6;
  float sr = 0.f, si = 0.f, ga = 0.f;
  for (int d = 0; d < 64; ++d) {
    float fr = ffr[bt * 64 + d], fi = ffi[bt * 64 + d];
    float wr = wsr[d * 64 + e], wi = wsi[d * 64 + e];
    sr += fr * wr - fi * wi;
    si += fr * wi + fi * wr;
    ga += fr * wg[d * 64 + e];
  }
  gq[gid] = 1.f / (1.f + __expf(-(ga + bg[e])));
  srcr[gid] = sr; srci[gid] = si;
}

// ---------------------------------------------------- h-scan over T, y in place
__global__ void hscan_kernel(float* __restrict__ er, float* __restrict__ ei,
                             const float* __restrict__ gq,
                             const float* __restrict__ srcr, const float* __restrict__ srci,
                             const float* __restrict__ hpr, const float* __restrict__ hpi,
                             const float* __restrict__ dt,
                             const float* __restrict__ lhr, const float* __restrict__ lhi,
                             float* __restrict__ dout) {
  size_t gid = (size_t)blockIdx.x * blockDim.x + threadIdx.x;  // 1,048,576
  int e = (int)(gid & 63); size_t bp = gid >> 6; int b = (int)(bp >> 12); int p = (int)(bp & 4095);
  float dts = dt[0];
  float lr = -fabsf(lhr[e]), li = lhi[e];
  float ea = __expf(lr * dts);
  float ar = ea * __cosf(li * dts), ai = ea * __sinf(li * dts);
  float nr = ar - 1.f, ni = ai;
  float den = 1.f / (lr * lr + li * li);
  float cr = (nr * lr + ni * li) * den, ci = (ni * lr - nr * li) * den;
  float sr = hpr[bp * 64 + e], si = hpi[bp * 64 + e];
  for (int t = 0; t < 8; ++t) {
    size_t o = ((size_t)(b * 8 + t) * HW + p) * 64 + e;
    int fo = (b * 8 + t) * 64 + e;
    float g = gq[fo];
    float fr = er[o] * g + srcr[fo] * (1.f - g);
    float fi = ei[o] * g + srci[fo] * (1.f - g);
    float ur = fr * cr - fi * ci, ui = fr * ci + fi * cr;
    float tr = ar * sr - ai * si + ur;
    float ti = ar * si + ai * sr + ui;
    sr = tr; si = ti;
    er[o] = sr; ei[o] = si;
  }
  dout[OUT_H_OFF + bp * 64 + e] = sr;
  dout[OUT_H_OFF + 1048576 + bp * 64 + e] = si;
}

// --------------------------------------- complex GEMM: z = y([m,e]) @ Edec, channel-last out
__global__ void dec_kernel(const float* __restrict__ yr, const float* __restrict__ yi,
                           const float* __restrict__ Er, const float* __restrict__ Ei,
                           float* __restrict__ z) {
  __shared__ float sEr[64][66];
  __shared__ float sEi[64][66];
  int tid = threadIdx.x; int lane = tid & 31; int w = tid >> 5;
  int row = lane & 15, kh = lane >> 4;
  for (int i = 0; i < 16; ++i) {
    int idx = i * 256 + tid;
    int d = idx >> 6, e = idx & 63;
    stage_f32(Er + idx, &sEr[e][d]);
    stage_f32(Ei + idx, &sEi[e][d]);
  }
  stage_fence();
  int nt = w >> 1;
  int col = nt * 16 + row;
  for (int s = 0; s < 4; ++s) {
    int mt = blockIdx.x * 8 + (w & 1) + s * 2;
    int gm0 = mt * 16;
    const float* arb = yr + (size_t)(gm0 + row) * 64;
    const float* aib = yi + (size_t)(gm0 + row) * 64;
    v8f P1 = v8zero(), P2 = v8zero(), P3 = v8zero(), P4 = v8zero();
    for (int k = 0; k < 64; k += 4) {
      int d0 = k + 2 * kh;
      v2f ar, ai, er, ei;
      ar.x = arb[d0];       ar.y = arb[d0 + 1];
      ai.x = aib[d0];       ai.y = aib[d0 + 1];
      er.x = sEr[col][d0];  er.y = sEr[col][d0 + 1];
      ei.x = sEi[col][d0];  ei.y = sEi[col][d0 + 1];
      P1 = wmma4(ar, er, P1);
      P2 = wmma4(ai, ei, P2);
      P3 = wmma4(ar, ei, P3);
      P4 = wmma4(ai, er, P4);
    }
    #pragma unroll
    for (int r = 0; r < 8; ++r) {
      int m = r + 8 * kh;
      z[(size_t)(gm0 + m) * 128 + col] = P1[r] - P2[r];        // real -> c
      z[(size_t)(gm0 + m) * 128 + 64 + col] = P3[r] + P4[r];   // imag -> c+64
    }
  }
}

// --------------------------------------- fused FFN (128->512->128) + residual epilogue
__global__ void ffn_kernel(const float* __restrict__ z, const float* __restrict__ w1,
                           const float* __restrict__ b1, const float* __restrict__ w2,
                           const float* __restrict__ b2, const float* __restrict__ xspr,
                           const float* __restrict__ xspi, float* __restrict__ out) {
  __shared__ float sh[8][16][17];
  __shared__ float sW1[16][130];
  __shared__ float sW2[128][18];
  int tid = threadIdx.x; int lane = tid & 31; int w = tid >> 5;
  int row = lane & 15, kh = lane >> 4;
  int mt = blockIdx.x * 8 + w;          // 8192 m-tiles total
  int gm0 = mt * 16;
  v8f acc[8];
  #pragma unroll
  for (int ct = 0; ct < 8; ++ct) acc[ct] = v8zero();

  for (int fc = 0; fc < 32; ++fc) {     // hidden chunks of 16
    // stage w1 chunk transposed: sW1[c][k] = w1[k*512 + fc*16 + c]
    for (int i = 0; i < 8; ++i) {
      int idx = i * 256 + tid;          // k*16 + c
      int k = idx >> 4, c = idx & 15;
      stage_f32(w1 + (size_t)k * 512 + fc * 16 + c, &sW1[c][k]);
    }
    // stage w2 chunk transposed: sW2[c][j] = w2[(fc*16+j)*128 + c]
    for (int i = 0; i < 8; ++i) {
      int idx = i * 256 + tid;          // j*128 + c
      int j = idx >> 7, c = idx & 127;
      stage_f32(w2 + (size_t)(fc * 16 + j) * 128 + c, &sW2[c][j]);
    }
    stage_fence();
    v8f h = v8zero();
    for (int k = 0; k < 128; k += 4) {
      int k0 = k + 2 * kh;
      v2f a, b;
      a.x = z[(size_t)(gm0 + row) * 128 + k0];     // contiguous -> b64
      a.y = z[(size_t)(gm0 + row) * 128 + k0 + 1];
      b.x = sW1[row][k0];                          // ds_load_b64
      b.y = sW1[row][k0 + 1];
      h = wmma4(a, b, h);
    }
    // bias + gelu(tanh), relayout D-layout -> A-fragment layout via per-wave LDS tile
    #pragma unroll
    for (int r = 0; r < 8; ++r) {
      int m = r + 8 * kh;
      float v = h[r] + b1[fc * 16 + row];
      float t = tanhf(0.7978845608028654f * (v + 0.044715f * v * v * v));
      sh[w][m][row] = 0.5f * v * (1.f + t);
    }
    __syncthreads();
    for (int kb = 0; kb < 16; kb += 4) {
      int j0 = kb + 2 * kh;
      v2f a2;
      a2.x = sh[w][row][j0];
      a2.y = sh[w][row][j0 + 1];
      #pragma unroll
      for (int ct = 0; ct < 8; ++ct) {
        v2f b2v;
        b2v.x = sW2[ct * 16 + row][j0];
        b2v.y = sW2[ct * 16 + row][j0 + 1];
        acc[ct] = wmma4(a2, b2v, acc[ct]);
      }
    }
    __syncthreads();
  }
  // epilogue: out = xsp + z + (o + b2), stacked real/imag planes in NCHW
  #pragma unroll
  for (int ct = 0; ct < 8; ++ct) {
    #pragma unroll
    for (int r = 0; r < 8; ++r) {
      int m = r + 8 * kh; int c = ct * 16 + row;
      int pm = gm0 + m; int bt = pm >> 12; int p = pm & 4095; int d = c & 63;
      float o = acc[ct][r] + b2[c];
      const float* xsp = (c < 64) ? xspr : xspi;
      float zv = z[(size_t)pm * 128 + c];
      float fin = xsp[(size_t)pm * 64 + d] + zv + o;
      out[(size_t)(c < 64 ? 0 : 1) * 8388608 + ((size_t)bt * 64 + d) * HW + p] = fin;
    }
  }
}

// ---------------------------------------------------------------- launcher
extern "C" void kernel_launch(void* const* d_in, const int* in_sizes, int n_in,
                              void* d_out, int out_size, void* d_ws, size_t ws_size,
                              hipStream_t stream) {
  const float* x_re   = (const float*)d_in[0];
  const float* x_im   = (const float*)d_in[1];
  const float* hpr    = (const float*)d_in[2];
  const float* hpi    = (const float*)d_in[3];
  const float* fpr    = (const float*)d_in[4];
  const float* fpi    = (const float*)d_in[5];
  const float* dt     = (const float*)d_in[6];
  const float* ln_g   = (const float*)d_in[7];
  const float* ln_b   = (const float*)d_in[8];
  const float* conv_w = (const float*)d_in[9];
  const float* conv_b = (const float*)d_in[10];
  const float* metric = (const float*)d_in[11];
  const float* E_re   = (const float*)d_in[12];
  const float* E_im   = (const float*)d_in[13];
  const float* Ed_re  = (const float*)d_in[14];
  const float* Ed_im  = (const float*)d_in[15];
  const float* lhr    = (const float*)d_in[16];
  const float* lhi    = (const float*)d_in[17];
  const float* lfr    = (const float*)d_in[18];
  const float* lfi    = (const float*)d_in[19];
  const float* wsr    = (const float*)d_in[20];
  const float* wsi    = (const float*)d_in[21];
  const float* wg     = (const float*)d_in[22];
  const float* bg     = (const float*)d_in[23];
  const float* ffw1   = (const float*)d_in[24];
  const float* ffb1   = (const float*)d_in[25];
  const float* ffw2   = (const float*)d_in[26];
  const float* ffb2   = (const float*)d_in[27];
  float* out = (float*)d_out;
  float* ws  = (float*)d_ws;

  // workspace layout (floats), with aliasing of consumed buffers
  float* xn   = ws;                       // 16,777,216
  float* xs   = ws + 16777216;            // 16,777,216
  float* xspr = ws + 33554432;            // 8,388,608 (channel-last [pm][d])
  float* xspi = ws + 41943040;            // 8,388,608
  float* eigr = xs;                       // alias: xs consumed before eig written
  float* eigi = xs + 8388608;
  float* zbuf = xn;                       // alias: xn consumed before z written
  float* sm   = ws + 50331648;            // small buffers
  float* mr = sm, *mi = sm + 2048;
  float* ffr = sm + 4096, *ffi = sm + 6144;
  float* gq  = sm + 8192;
  float* srr = sm + 10240, *sri = sm + 12288;
  float* cwt = sm + 16384;                // 147,456 repacked conv weights

  ln_kernel<<<512, 256, 0, stream>>>(x_re, x_im, ln_g, ln_b, xn);
  repack_kernel<<<576, 256, 0, stream>>>(conv_w, cwt);
  conv_kernel<<<2048, 256, 0, stream>>>(xn, cwt, xs);
  xsp_kernel<<<32768, 256, 0, stream>>>(x_re, x_im, xs, conv_b, metric, xspr, xspi);
  eig_kernel<<<1024, 256, 0, stream>>>(xspr, xspi, E_re, E_im, eigr, eigi);
  mean_kernel<<<32, 256, 0, stream>>>(eigr, eigi, mr, mi);
  flux_kernel<<<1, 256, 0, stream>>>(mr, mi, fpr, fpi, dt, lfr, lfi, ffr, ffi, out);
  srcgate_kernel<<<8, 256, 0, stream>>>(ffr, ffi, wsr, wsi, wg, bg, gq, srr, sri);
  hscan_kernel<<<4096, 256, 0, stream>>>(eigr, eigi, gq, srr, sri, hpr, hpi, dt, lhr, lhi, out);
  dec_kernel<<<1024, 256, 0, stream>>>(eigr, eigi, Ed_re, Ed_im, zbuf);
  ffn_kernel<<<1024, 256, 0, stream>>>(zbuf, ffw1, ffb1, ffw2, ffb2, xspr, xspi, out);
}